// EquivariantLayerNorm_72541997630029
// MI455X (gfx1250) — compile-verified
//
#include <hip/hip_runtime.h>
#include <hip/hip_bf16.h>
#include <stdint.h>

// ---- problem constants (from IRREPS = [(256,0),(128,1),(64,2),(32,3)]) ----
#define DIM      1184     // 256*1 + 128*3 + 64*5 + 32*7
#define NCHUNK   9        // 9 float4 chunks per lane cover cols [0,1152)
#define TAILBASE 1152     // cols [1152,1184): one scalar per lane
#define BLOCK    256      // 8 waves of 32
#define WPB      8
#define GRID     1024     // 8192 waves -> 8 rows per wave at 65536 rows
#define EPS_F    1e-5f

typedef float f4  __attribute__((ext_vector_type(4)));
typedef int   v4i __attribute__((ext_vector_type(4)));

// CDNA5 async global->LDS DMA path (ASYNCcnt), guarded so we compile either way.
#if defined(__HIP_DEVICE_COMPILE__) && defined(__gfx1250__) && \
    __has_builtin(__builtin_amdgcn_global_load_async_to_lds_b128) && \
    __has_builtin(__builtin_amdgcn_global_load_async_to_lds_b32)
#define HAVE_ASYNC 1
#else
#define HAVE_ASYNC 0
#endif

#if HAVE_ASYNC
#define AS1 __attribute__((address_space(1)))
#define AS3 __attribute__((address_space(3)))
// Generic LDS pointer low 32 bits == LDS byte offset (aperture mapping), so an
// integer round-trip safely materializes addrspace-qualified typed pointers.
#define GLB_PTR(T, p) ((AS1 T*)(uintptr_t)(p))
#define LDS_PTR(T, p) ((AS3 T*)(uint32_t)(uintptr_t)(p))

// One row = 9 x b128 (512B/wave each) + 1 x b32 (128B/wave) = 4736 B = 10 async ops.
__device__ __forceinline__ void async_row_load(const float* __restrict__ src,
                                               float* dst, int lane) {
  const char* g = (const char*)src;
  const char* l = (const char*)dst;
#pragma unroll
  for (int k = 0; k < NCHUNK; ++k) {
    const int off = k * 512 + lane * 16;
    __builtin_amdgcn_global_load_async_to_lds_b128(GLB_PTR(v4i, g + off),
                                                   LDS_PTR(v4i, l + off), 0, 0);
  }
  const int toff = TAILBASE * 4 + lane * 4;
  __builtin_amdgcn_global_load_async_to_lds_b32(GLB_PTR(int, g + toff),
                                                LDS_PTR(int, l + toff), 0, 0);
}
#endif

__device__ __forceinline__ float hsum4(f4 a)  { return (a.x + a.y) + (a.z + a.w); }
__device__ __forceinline__ float sumsq4(f4 a) { return a.x*a.x + a.y*a.y + a.z*a.z + a.w*a.w; }

__launch_bounds__(BLOCK)
__global__ void eqln_kernel(const float* __restrict__ x,
                            const float* __restrict__ weight,
                            const float* __restrict__ bias,
                            const int*   __restrict__ irrep_idx,
                            float* __restrict__ out, int rows) {
#if HAVE_ASYNC
  extern __shared__ float ldsbuf[];   // WPB waves * 2 buffers * DIM floats
#endif
  const int lane = threadIdx.x & 31;
  const int wib  = threadIdx.x >> 5;
  const int wgid = blockIdx.x * WPB + wib;
  const int tw   = gridDim.x * WPB;

  // Per-column constants, loaded once per wave, reused for all its rows.
  // scale[c] = weight[irrep_idx[c]]; bias applies to cols [0,256) (chunks 0,1).
  f4 scl[NCHUNK]; f4 bia[2]; float sclT;
#pragma unroll
  for (int k = 0; k < NCHUNK; ++k) {
    const int c = k * 128 + lane * 4;
    const int4 ii = *(const int4*)(irrep_idx + c);
    f4 s;
    s.x = weight[ii.x]; s.y = weight[ii.y]; s.z = weight[ii.z]; s.w = weight[ii.w];
    scl[k] = s;
  }
  sclT   = weight[irrep_idx[TAILBASE + lane]];
  bia[0] = *(const f4*)(bias + lane * 4);
  bia[1] = *(const f4*)(bias + 128 + lane * 4);

  int r = wgid;
  if (r >= rows) return;

#if HAVE_ASYNC
  float* buf0 = ldsbuf + (wib * 2 + 0) * DIM;
  float* buf1 = ldsbuf + (wib * 2 + 1) * DIM;
  async_row_load(x + (size_t)r * DIM, buf0, lane);
  int sel = 0;
#endif

  for (;;) {
    const int  nr       = r + tw;
    const bool has_next = (nr < rows);
#if HAVE_ASYNC
    if (has_next) async_row_load(x + (size_t)nr * DIM, sel ? buf0 : buf1, lane);
    // Async loads complete in order: <=10 outstanding means row r's 10 are done.
    if (has_next) asm volatile("s_wait_asynccnt 10" ::: "memory");
    else          asm volatile("s_wait_asynccnt 0"  ::: "memory");
    const float* vsrc = sel ? buf1 : buf0;
#else
    const float* vsrc = x + (size_t)r * DIM;
#endif

    f4 v[NCHUNK]; float vt;
#pragma unroll
    for (int k = 0; k < NCHUNK; ++k)
      v[k] = *(const f4*)(vsrc + k * 128 + lane * 4);
    vt = vsrc[TAILBASE + lane];

    // Group membership per chunk (c = 128k + 4*lane + j):
    //   k=0,1 -> g0 [0,256); k=2..4 -> g1 [256,640); k=5,6 -> g2;
    //   k=7 -> g2 if lane<16 else g3 (boundary 960); k=8 + tail -> g3.
    float ssum = hsum4(v[0]) + hsum4(v[1]);
    float sq0  = sumsq4(v[0]) + sumsq4(v[1]);
    float sq1  = sumsq4(v[2]) + sumsq4(v[3]) + sumsq4(v[4]);
    float sq2  = sumsq4(v[5]) + sumsq4(v[6]);
    float sq3  = sumsq4(v[8]) + vt * vt;
    {
      const float s7 = sumsq4(v[7]);
      if (lane < 16) sq2 += s7; else sq3 += s7;
    }

    // wave32 butterfly reduction
#pragma unroll
    for (int off = 16; off > 0; off >>= 1) {
      ssum += __shfl_xor(ssum, off, 32);
      sq0  += __shfl_xor(sq0,  off, 32);
      sq1  += __shfl_xor(sq1,  off, 32);
      sq2  += __shfl_xor(sq2,  off, 32);
      sq3  += __shfl_xor(sq3,  off, 32);
    }

    const float m   = ssum * (1.0f / 256.0f);
    const float rn0 = rsqrtf(fmaxf(sq0 * (1.0f / 256.0f) - m * m, 0.0f) + EPS_F);
    const float rn1 = rsqrtf(sq1 * (1.0f / 384.0f) + EPS_F);
    const float rn2 = rsqrtf(sq2 * (1.0f / 320.0f) + EPS_F);
    const float rn3 = rsqrtf(sq3 * (1.0f / 224.0f) + EPS_F);
    const float rn7 = (lane < 16) ? rn2 : rn3;

    float* orow = out + (size_t)r * DIM;
#pragma unroll
    for (int k = 0; k < NCHUNK; ++k) {
      const float rk = (k < 2) ? rn0 : (k < 5) ? rn1 : (k < 7) ? rn2
                     : (k == 7) ? rn7 : rn3;
      f4 o;
      if (k < 2) o = (v[k] - m) * rk * scl[k] + bia[k];
      else       o = v[k] * rk * scl[k];
      __builtin_nontemporal_store(o, (f4*)(orow + k * 128 + lane * 4));
    }
    __builtin_nontemporal_store(vt * rn3 * sclT, orow + TAILBASE + lane);

    if (!has_next) break;
    r = nr;
#if HAVE_ASYNC
    sel ^= 1;
#endif
  }
}

extern "C" void kernel_launch(void* const* d_in, const int* in_sizes, int n_in,
                              void* d_out, int out_size, void* d_ws, size_t ws_size,
                              hipStream_t stream) {
  // setup_inputs order: x, weight, bias, group_idx, irrep_idx, scalar_indices, scalar_group
  const float* x      = (const float*)d_in[0];
  const float* weight = (const float*)d_in[1];
  const float* bias   = (const float*)d_in[2];
  const int*   irrep  = (const int*)d_in[4];
  const int rows = in_sizes[0] / DIM;
  const size_t shmem = (size_t)WPB * 2 * DIM * sizeof(float); // 75,776 B / block
  eqln_kernel<<<GRID, BLOCK, shmem, stream>>>(x, weight, bias, irrep,
                                              (float*)d_out, rows);
}